// GNN_24386824306931
// MI455X (gfx1250) — compile-verified
//
#include <hip/hip_runtime.h>
#include <hip/hip_bf16.h>
#include <math.h>

// ---------------------------------------------------------------------------
// GNN on MI455X (gfx1250, wave32).
//   agg[d] = sum_e Y[type[e]][src[e]],  Y_t = X @ P_t  (8 dense WMMA GEMMs)
//   out = (out + agg) * 0.5, 3 layers; then normalize + MLP (WMMA) + sigmoid.
// GEMMs: V_WMMA_F32_16X16X4_F32, B panels staged in LDS via the gfx1250
// async global->LDS path (ASYNCcnt), A fragments held in VGPRs.
// ---------------------------------------------------------------------------

typedef float v2f __attribute__((ext_vector_type(2)));
typedef float v8f __attribute__((ext_vector_type(8)));

#define HDIM 32
#define WAVES_PER_BLOCK 8

// gfx1250 async copy: global -> LDS, 4 bytes per lane. Tracked by ASYNCcnt.
__device__ __forceinline__ void async_copy_b32(unsigned ldsByteOff,
                                               const float* gsrc) {
  asm volatile("global_load_async_to_lds_b32 %0, %1, off"
               :: "v"(ldsByteOff), "v"(gsrc)
               : "memory");
}
__device__ __forceinline__ void wait_async0() {
  asm volatile("s_wait_asynccnt 0x0" ::: "memory");
}

// ---- init: Xcur = feature, agg = 0 ----------------------------------------
__global__ void init_kernel(const float* __restrict__ feat,
                            float* __restrict__ x,
                            float* __restrict__ agg, int n) {
  int i = blockIdx.x * blockDim.x + threadIdx.x;
  if (i < n) { x[i] = feat[i]; agg[i] = 0.0f; }
}

// ---- batched WMMA f32 GEMM: C_b = act(A @ B_b + bias), b = 0..NB-1 --------
// A: M x K row-major (shared across batches), B_b: K x (NT*16) row-major,
// C_b: M x (NT*16) row-major.  One wave owns one 16-row stripe and computes
// NB * NT 16x16 tiles with V_WMMA_F32_16X16X4_F32 (K/4 steps each).
// A 16x4 f32 lane layout: M = lane%16; lane half selects K pair base (0 / 2),
// the two A VGPRs hold K = ka, ka+1.  B 4x16: VGPR0/1 = rows ka, ka+1, lane
// half selects ka base.  D: VGPR r -> row r (lanes 0-15) / r+8 (lanes 16-31).
template <int K, int NT, int NB>
__global__ void gemm_bias_act_wmma(const float* __restrict__ A,
                                   const float* __restrict__ B, long strideB,
                                   const float* __restrict__ bias,
                                   float* __restrict__ C, long strideC,
                                   int M, int act /*0=none,1=tanh*/) {
  constexpr int NCOLS  = NT * 16;
  constexpr int BPANEL = K * NCOLS;          // floats per batch panel
  __shared__ float Bsh[NB * BPANEL];         // <= 32 KB (WGP has 320 KB)

  const int tid = threadIdx.x;

  // Stage all NB weight panels into LDS with async global->LDS copies.
  for (int i = tid; i < NB * BPANEL; i += (int)blockDim.x) {
    const int b = i / BPANEL;
    const int r = i - b * BPANEL;
    async_copy_b32((unsigned)(unsigned long)(&Bsh[i]),
                   B + (long)b * strideB + r);
  }
  wait_async0();          // drain this wave's ASYNCcnt
  __syncthreads();        // make LDS panel visible block-wide

  const int wave   = tid >> 5;
  const int lane   = tid & 31;
  const int mtile  = blockIdx.x * WAVES_PER_BLOCK + wave;
  const int mtiles = M >> 4;
  if (mtile < mtiles) {                      // wave-uniform: EXEC stays all-1s
    const int half2 = (lane >> 4) << 1;      // K sub-base: 0 or 2
    const int colL  = lane & 15;
    const int mrowA = (mtile << 4) + colL;   // A row (M = lane%16)
    constexpr int NK = K >> 2;

    // Pre-load this wave's A fragments once; reused for all NB*NT tiles.
    v2f afrag[NK];
#pragma unroll
    for (int kk = 0; kk < NK; ++kk) {
      const int ka = (kk << 2) + half2;
      afrag[kk][0] = A[(long)mrowA * K + ka];   // contiguous pair -> b64 load
      afrag[kk][1] = A[(long)mrowA * K + ka + 1];
    }

    const int rbase = (mtile << 4) + ((lane >> 4) << 3);
#pragma unroll 1
    for (int b = 0; b < NB; ++b) {
      const float* Bp = &Bsh[b * BPANEL];
      float*       Cb = C + (long)b * strideC;
#pragma unroll
      for (int nt = 0; nt < NT; ++nt) {
        const int col = (nt << 4) + colL;
        v8f acc = {0.f, 0.f, 0.f, 0.f, 0.f, 0.f, 0.f, 0.f};
#pragma unroll
        for (int kk = 0; kk < NK; ++kk) {
          const int ka = (kk << 2) + half2;
          v2f bm;
          bm[0] = Bp[ka * NCOLS + col];        // conflict-free ds_load
          bm[1] = Bp[(ka + 1) * NCOLS + col];
          acc = __builtin_amdgcn_wmma_f32_16x16x4_f32(
              /*neg_a=*/false, afrag[kk], /*neg_b=*/false, bm,
              /*c_mod=*/(short)0, acc, /*reuse_a=*/false, /*reuse_b=*/false);
        }
        const float bc = bias ? bias[col] : 0.0f;
#pragma unroll
        for (int r = 0; r < 8; ++r) {
          float v = acc[r] + bc;
          if (act) v = tanhf(v);
          Cb[(long)(rbase + r) * NCOLS + col] = v;
        }
      }
    }
  }
}

// ---- edge pass: agg[dst] += Y[type][src]  (gather + f32 atomic scatter) ---
__global__ void edge_scatter(const float* __restrict__ Y,
                             const int* __restrict__ src,
                             const int* __restrict__ dst,
                             const int* __restrict__ etype,
                             float* __restrict__ agg,
                             int nEdges, int nNodes) {
  int gid = blockIdx.x * blockDim.x + threadIdx.x;
  int e = gid >> 3;                 // 8 threads per edge, 4 floats each
  if (e >= nEdges) return;
  int c = (gid & 7) << 2;

  // prefetch a future edge's source row into cache (global_prefetch_b8)
  int ep = e + 256;
  if (ep < nEdges) {
    long prow = ((long)etype[ep] * nNodes + src[ep]) * HDIM;
    __builtin_prefetch(Y + prow, 0, 0);
  }

  long srow = ((long)etype[e] * nNodes + src[e]) * HDIM + c;
  float4 v = *reinterpret_cast<const float4*>(Y + srow);
  float* p = agg + (long)dst[e] * HDIM + c;
  atomicAdd(p + 0, v.x);            // global_atomic_add_f32, no return
  atomicAdd(p + 1, v.y);
  atomicAdd(p + 2, v.z);
  atomicAdd(p + 3, v.w);
}

// ---- combine: x = (x + agg)*0.5, re-zero agg for next layer ---------------
__global__ void combine_kernel(float* __restrict__ x, float* __restrict__ agg,
                               int n) {
  int i = blockIdx.x * blockDim.x + threadIdx.x;
  if (i < n) { x[i] = (x[i] + agg[i]) * 0.5f; agg[i] = 0.0f; }
}

// ---- per-node L2 normalize ------------------------------------------------
__global__ void normalize_kernel(const float* __restrict__ x,
                                 float* __restrict__ xn, int nNodes) {
  int nidx = blockIdx.x * blockDim.x + threadIdx.x;
  if (nidx >= nNodes) return;
  const float4* row = reinterpret_cast<const float4*>(x + (long)nidx * HDIM);
  float4 v[8];
  float ss = 0.0f;
#pragma unroll
  for (int i = 0; i < 8; ++i) {
    v[i] = row[i];
    ss += v[i].x * v[i].x + v[i].y * v[i].y + v[i].z * v[i].z + v[i].w * v[i].w;
  }
  float inv = 1.0f / fmaxf(sqrtf(ss), 1e-12f);
  float4* o = reinterpret_cast<float4*>(xn + (long)nidx * HDIM);
#pragma unroll
  for (int i = 0; i < 8; ++i) {
    float4 w = v[i];
    w.x *= inv; w.y *= inv; w.z *= inv; w.w *= inv;
    o[i] = w;
  }
}

// ---- head: sigmoid(h2 @ w3 + b3) ------------------------------------------
__global__ void head_kernel(const float* __restrict__ h2,
                            const float* __restrict__ w3,
                            const float* __restrict__ b3,
                            float* __restrict__ out, int nNodes) {
  int nidx = blockIdx.x * blockDim.x + threadIdx.x;
  if (nidx >= nNodes) return;
  const float* r = h2 + (long)nidx * 64;
  float s = b3[0];
#pragma unroll
  for (int i = 0; i < 64; ++i) s += r[i] * w3[i];
  out[nidx] = 1.0f / (1.0f + __expf(-s));
}

// ---------------------------------------------------------------------------
extern "C" void kernel_launch(void* const* d_in, const int* in_sizes, int n_in,
                              void* d_out, int out_size, void* d_ws,
                              size_t ws_size, hipStream_t stream) {
  const float* feature = (const float*)d_in[0];
  const float* param   = (const float*)d_in[1];
  const float* w1      = (const float*)d_in[2];
  const float* b1      = (const float*)d_in[3];
  const float* w2      = (const float*)d_in[4];
  const float* b2      = (const float*)d_in[5];
  const float* w3      = (const float*)d_in[6];
  const float* b3      = (const float*)d_in[7];
  const int*   src     = (const int*)d_in[8];
  const int*   dst     = (const int*)d_in[9];
  const int*   pid     = (const int*)d_in[10];

  const int N = in_sizes[0] / HDIM;                 // 50000 (multiple of 16)
  const int E = in_sizes[8];                        // 400000
  const int nfeat = N * HDIM;

  // Workspace layout (floats). Post-loop buffers alias the Y region
  // (Y = 8*N*32 floats >= N*32 + N*64 + N*64).
  float* ws   = (float*)d_ws;
  float* Xcur = ws;
  float* Agg  = ws + (size_t)nfeat;
  float* Y    = ws + (size_t)2 * nfeat;
  float* Xn   = Y;
  float* H1   = Y + (size_t)nfeat;
  float* H2   = H1 + (size_t)N * 64;

  init_kernel<<<(nfeat + 255) / 256, 256, 0, stream>>>(feature, Xcur, Agg,
                                                       nfeat);

  const int mtiles  = N / 16;
  const int gBlocks = (mtiles + WAVES_PER_BLOCK - 1) / WAVES_PER_BLOCK;
  const int eThreads = E * 8;

  for (int layer = 0; layer < 3; ++layer) {
    // Y_t = Xcur @ P_t for all 8 edge types; all 8 panels staged in LDS once.
    gemm_bias_act_wmma<32, 2, 8><<<gBlocks, 256, 0, stream>>>(
        Xcur, param, (long)HDIM * HDIM, nullptr, Y, (long)N * HDIM, N, 0);
    // agg[dst] += Y[type][src]
    edge_scatter<<<(eThreads + 255) / 256, 256, 0, stream>>>(
        Y, src, dst, pid, Agg, E, N);
    // Xcur = (Xcur + agg)*0.5, agg := 0
    combine_kernel<<<(nfeat + 255) / 256, 256, 0, stream>>>(Xcur, Agg, nfeat);
  }

  normalize_kernel<<<(N + 255) / 256, 256, 0, stream>>>(Xcur, Xn, N);

  // MLP: H1 = tanh(Xn @ w1 + b1); H2 = tanh(H1 @ w2 + b2)
  gemm_bias_act_wmma<32, 4, 1><<<gBlocks, 256, 0, stream>>>(
      Xn, w1, 0L, b1, H1, 0L, N, 1);
  gemm_bias_act_wmma<64, 4, 1><<<gBlocks, 256, 0, stream>>>(
      H1, w2, 0L, b2, H2, 0L, N, 1);

  head_kernel<<<(N + 255) / 256, 256, 0, stream>>>(H2, w3, b3, (float*)d_out,
                                                   N);
}